// CrfDecodeLayer_72395968741564
// MI455X (gfx1250) — compile-verified
//
#include <hip/hip_runtime.h>
#include <stdint.h>
#include <stddef.h>

#define BATCH 256
#define TLEN  512
#define KK    128
#define PADSTRIDE 132   // 128 dwords + 4 dwords TDM pad -> conflict-free LDS columns

typedef unsigned int uint32x4_t __attribute__((ext_vector_type(4)));
typedef int          int32x4_t  __attribute__((ext_vector_type(4)));
typedef int          int32x8_t  __attribute__((ext_vector_type(8)));

#ifndef __has_builtin
#define __has_builtin(x) 0
#endif

#if __has_builtin(__builtin_amdgcn_tensor_load_to_lds)
#define HAVE_TDM 1
#else
#define HAVE_TDM 0
#endif

__device__ __forceinline__ void wait_tensorcnt0() {
#if __has_builtin(__builtin_amdgcn_s_wait_tensorcnt)
  __builtin_amdgcn_s_wait_tensorcnt(0);
#else
  asm volatile("s_wait_tensorcnt 0x0" ::: "memory");
#endif
}

// LDS layout (dynamic):
//   [0)                trans_p : KK*PADSTRIDE floats (67584 B), row-major padded
//   [67584)            sbuf    : 2*KK floats (double-buffered Viterbi state)
//   [68608)            tags    : TLEN ints
//   [70656)            red     : KK ints (length reduction)
#define SMEM_BYTES (KK*PADSTRIDE*4 + 2*KK*4 + TLEN*4 + KK*4)   // 71168

__global__ __launch_bounds__(KK) void crf_viterbi_kernel(
    const float* __restrict__ emissions,     // [B,T,K] f32
    const int*   __restrict__ mask,          // [B,T] i32 (0/1)
    const float* __restrict__ trans,         // [K,K] f32
    int*         __restrict__ out,           // [B,T] i32
    uint32_t*    __restrict__ bp_ws)         // [B, T/4, K] packed backpointers
{
  extern __shared__ char smem[];
  float* trans_p = (float*)smem;
  float* sbuf    = (float*)(smem + KK*PADSTRIDE*4);
  int*   tags    = (int*)  (smem + KK*PADSTRIDE*4 + 2*KK*4);
  int*   red     = (int*)  (smem + KK*PADSTRIDE*4 + 2*KK*4 + TLEN*4);

  const int tid = threadIdx.x;
  const int b   = blockIdx.x;
  const float* em  = emissions + (size_t)b * TLEN * KK;
  const int*   mk  = mask      + (size_t)b * TLEN;
  uint32_t*    wsb = bp_ws     + (size_t)b * (TLEN/4) * KK;

  // ---- Load transition matrix into LDS with padded stride ----
#if HAVE_TDM
  if (tid < 32) {  // wave 0 issues one TDM op; EXEC ignored by TDM
    const unsigned long long ga = (unsigned long long)(uintptr_t)trans;
    const unsigned lds_addr = (unsigned)(uintptr_t)trans_p;  // low 32 bits = LDS byte addr
    uint32x4_t g0;
    g0[0] = 1u;                                   // count=1, user descriptor
    g0[1] = lds_addr;
    g0[2] = (unsigned)(ga & 0xFFFFFFFFull);       // global_addr[31:0]
    g0[3] = (unsigned)((ga >> 32) & 0x1FFFFFFull) | (2u << 30);  // addr[56:32] | type=2
    int32x8_t g1;
    // data_size=4B (2<<16), pad_enable (1<<20), pad_interval=128dw (6<<22), pad_amount=4dw (3<<25)
    g1[0] = (int)((2u << 16) | (1u << 20) | (6u << 22) | (3u << 25));
    g1[1] = (int)(128u << 16);   // tensor_dim0 = 128 (bits [79:48])
    g1[2] = (int)(128u << 16);   // tensor_dim1 = 128 (bits [111:80])
    g1[3] = (int)(128u << 16);   // tile_dim0   = 128 (bits [127:112])
    g1[4] = 128;                 // tile_dim1   = 128, tile_dim2 = 0
    g1[5] = 128;                 // tensor_dim0_stride = 128 (low 32 of 48)
    g1[6] = 0;
    g1[7] = 0;
    int32x4_t z4 = {0, 0, 0, 0};
#if __clang_major__ >= 23
    int32x8_t z8 = {0, 0, 0, 0, 0, 0, 0, 0};
    __builtin_amdgcn_tensor_load_to_lds(g0, g1, z4, z4, z8, 0);
#else
    __builtin_amdgcn_tensor_load_to_lds(g0, g1, z4, z4, 0);
#endif
    wait_tensorcnt0();
  }
#else
  for (int idx = tid; idx < KK * KK; idx += KK) {
    int i = idx >> 7, j = idx & (KK - 1);
    trans_p[i * PADSTRIDE + j] = trans[idx];
  }
#endif

  // ---- sequence length (uniform per batch) + state init ----
  int part = 0;
  for (int t = tid; t < TLEN; t += KK) part += mk[t];
  red[tid]  = part;
  sbuf[tid] = em[tid];          // init_state = emissions[:,0,:]
  __syncthreads();              // also fences TDM result (wave0 waited tensorcnt)
  if (tid == 0) {
    int s = 0;
    #pragma unroll
    for (int i = 0; i < KK; ++i) s += red[i];
    red[0] = s;
  }
  __syncthreads();
  const int len = red[0];

  // ---- forward Viterbi recurrence ----
  uint32_t packed = 0;
  int t = 1;
  for (; t < len; ++t) {        // state frozen for t >= len; those bps are never read
    const float* sold = sbuf + ((t - 1) & 1) * KK;
    float*       snew = sbuf + (t & 1) * KK;
    const float  e    = em[t * KK + tid];
    if (t + 1 < len) __builtin_prefetch(&em[(t + 1) * KK + tid], 0, 1);

    float best = -3.402823466e38f;
    int   bpj  = 0;
    const float* tcol = trans_p + tid;   // column tid, stride PADSTRIDE (bank-conflict-free)
    #pragma unroll 8
    for (int i = 0; i < KK; i += 4) {
      const float4 st = *(const float4*)(sold + i);   // uniform ds_load_b128 broadcast
      float v0 = st.x + tcol[(i + 0) * PADSTRIDE];
      float v1 = st.y + tcol[(i + 1) * PADSTRIDE];
      float v2 = st.z + tcol[(i + 2) * PADSTRIDE];
      float v3 = st.w + tcol[(i + 3) * PADSTRIDE];
      if (v0 > best) { best = v0; bpj = i;     }      // strict > == jnp.argmax first-max
      if (v1 > best) { best = v1; bpj = i + 1; }
      if (v2 > best) { best = v2; bpj = i + 2; }
      if (v3 > best) { best = v3; bpj = i + 3; }
    }
    snew[tid] = best + e;
    packed |= (uint32_t)bpj << ((t & 3) * 8);
    if ((t & 3) == 3) { wsb[(t >> 2) * KK + tid] = packed; packed = 0; }
    __syncthreads();
  }
  if (t > 1 && ((t - 1) & 3) != 3)                    // flush partial packed word
    wsb[((t - 1) >> 2) * KK + tid] = packed;

  __threadfence_block();
  __syncthreads();

  // ---- backward chase (serial per batch) ----
  if (tid == 0) {
    const float* fs = sbuf + ((len > 1) ? ((len - 1) & 1) : 0) * KK;
    float best = fs[0];
    int   tag  = 0;
    for (int i = 1; i < KK; ++i) { float v = fs[i]; if (v > best) { best = v; tag = i; } }
    tags[TLEN - 1] = tag;
    for (int tt = TLEN - 1; tt >= 1; --tt) {
      if (tt < len) {
        uint32_t w = wsb[(tt >> 2) * KK + tag];
        tag = (int)((w >> ((tt & 3) * 8)) & 0xFFu);
      }
      tags[tt - 1] = tag;
    }
  }
  __syncthreads();

  #pragma unroll
  for (int tt = tid; tt < TLEN; tt += KK)
    out[(size_t)b * TLEN + tt] = tags[tt] * mk[tt];
}

extern "C" void kernel_launch(void* const* d_in, const int* in_sizes, int n_in,
                              void* d_out, int out_size, void* d_ws, size_t ws_size,
                              hipStream_t stream) {
  (void)in_sizes; (void)n_in; (void)out_size; (void)ws_size;
  const float* emissions = (const float*)d_in[0];   // [256,512,128] f32
  const int*   mask      = (const int*)  d_in[1];   // [256,512] i32
  const float* trans     = (const float*)d_in[2];   // [128,128] f32
  int*         out       = (int*)d_out;             // [256,512] i32
  uint32_t*    bp_ws     = (uint32_t*)d_ws;         // needs 256*128*128*4 = 16 MB

  (void)hipFuncSetAttribute((const void*)crf_viterbi_kernel,
                            hipFuncAttributeMaxDynamicSharedMemorySize, SMEM_BYTES);

  crf_viterbi_kernel<<<BATCH, KK, SMEM_BYTES, stream>>>(emissions, mask, trans, out, bp_ws);
}